// mymodel_19327352832016
// MI455X (gfx1250) — compile-verified
//
#include <hip/hip_runtime.h>
#include <hip/hip_bf16.h>

// GNN-style sparse aggregation for MI455X (gfx1250, wave32):
//   out[i,:] = relu( sum_{e: row[e]==i} weight[col[e],:] + bias )
// N=50000 nodes, E=1.6M edges, C=32 channels.
//
// Bottleneck analysis: 51.2M f32 atomic adds into a 6.4MB output (L2-resident)
// + 205MB of weight-row gathers served from the 192MB L2 (weight table is only
// 6.4MB). WMMA is inapplicable (adjacency density 6.4e-4 -> ~0.16 nnz per
// 16x16 tile). We instead exercise the CDNA5-specific data-movement paths:
//  - GLOBAL_LOAD_ASYNC_TO_LDS_B32 (ASYNCcnt) to stage edge indices in LDS
//  - s_wait_asynccnt split-counter waits
//  - b128 vector loads of weight rows (128 contiguous bytes per lane)
//  - native no-return global_atomic_add_f32 at agent scope (relaxed)

#define C_CH 32
#define BLK  256

#ifndef USE_ASYNC_LDS
#define USE_ASYNC_LDS 1
#endif

__global__ __launch_bounds__(BLK) void zero_kernel(float4* __restrict__ out, int n4) {
    int i = blockIdx.x * BLK + threadIdx.x;
    if (i < n4) out[i] = make_float4(0.f, 0.f, 0.f, 0.f);
}

__global__ __launch_bounds__(BLK) void scatter_kernel(const int* __restrict__ row,
                                                      const int* __restrict__ col,
                                                      const float* __restrict__ weight,
                                                      float* __restrict__ out,
                                                      int n_edges) {
    const int eid   = blockIdx.x * BLK + threadIdx.x;
    const int esafe = (eid < n_edges) ? eid : (n_edges - 1);  // clamp so every lane has a valid addr

    int r, c;
#if USE_ASYNC_LDS
    // Stage this block's 256 row/col indices into LDS via the CDNA5 async
    // global->LDS DMA path. Each lane supplies its own global address and its
    // own LDS byte offset (low 32 bits of the flat shared-pointer = LDS addr,
    // per the gfx1250 LDS aperture rule). Completion is tracked by ASYNCcnt.
    __shared__ int s_row[BLK];
    __shared__ int s_col[BLK];
    {
        unsigned lds_r = (unsigned)(size_t)&s_row[threadIdx.x];
        unsigned lds_c = (unsigned)(size_t)&s_col[threadIdx.x];
        const int* gaddr_r = row + esafe;
        const int* gaddr_c = col + esafe;
        asm volatile("global_load_async_to_lds_b32 %0, %1, off"
                     :: "v"(lds_r), "v"(gaddr_r) : "memory");
        asm volatile("global_load_async_to_lds_b32 %0, %1, off"
                     :: "v"(lds_c), "v"(gaddr_c) : "memory");
        asm volatile("s_wait_asynccnt 0x0" ::: "memory");
    }
    r = s_row[threadIdx.x];
    c = s_col[threadIdx.x];
#else
    // Fallback: indices are streamed exactly once -> non-temporal loads keep
    // them from polluting caches that should hold the weight table.
    r = __builtin_nontemporal_load(row + esafe);
    c = __builtin_nontemporal_load(col + esafe);
#endif

    if (eid < n_edges) {
        // Gather the 32-float (128B) weight row: 8x global_load_b128 per lane,
        // each lane touching one full cacheline. Weight table (6.4MB) stays
        // L2-resident on the 192MB L2, so this is L2-bandwidth traffic.
        const float4* __restrict__ wrow = reinterpret_cast<const float4*>(weight + (size_t)c * C_CH);
        float4 w[8];
#pragma unroll
        for (int q = 0; q < 8; ++q) w[q] = wrow[q];

        // Scatter-accumulate: 32 contiguous f32 atomics per edge, relaxed
        // agent scope -> no-return global_atomic_add_f32 handled in L2.
        float* __restrict__ dst = out + (size_t)r * C_CH;
#pragma unroll
        for (int q = 0; q < 8; ++q) {
            __hip_atomic_fetch_add(dst + 4 * q + 0, w[q].x, __ATOMIC_RELAXED, __HIP_MEMORY_SCOPE_AGENT);
            __hip_atomic_fetch_add(dst + 4 * q + 1, w[q].y, __ATOMIC_RELAXED, __HIP_MEMORY_SCOPE_AGENT);
            __hip_atomic_fetch_add(dst + 4 * q + 2, w[q].z, __ATOMIC_RELAXED, __HIP_MEMORY_SCOPE_AGENT);
            __hip_atomic_fetch_add(dst + 4 * q + 3, w[q].w, __ATOMIC_RELAXED, __HIP_MEMORY_SCOPE_AGENT);
        }
    }
}

__global__ __launch_bounds__(BLK) void bias_relu_kernel(float4* __restrict__ out,
                                                        const float* __restrict__ bias,
                                                        int n4) {
    int i = blockIdx.x * BLK + threadIdx.x;
    if (i < n4) {
        const int q = i & 7;  // which float4 of the 32-channel row
        float4 v = out[i];
        const float4 b = reinterpret_cast<const float4*>(bias)[q];
        v.x = fmaxf(v.x + b.x, 0.f);
        v.y = fmaxf(v.y + b.y, 0.f);
        v.z = fmaxf(v.z + b.z, 0.f);
        v.w = fmaxf(v.w + b.w, 0.f);
        out[i] = v;
    }
}

extern "C" void kernel_launch(void* const* d_in, const int* in_sizes, int n_in,
                              void* d_out, int out_size, void* d_ws, size_t ws_size,
                              hipStream_t stream) {
    const int*   row    = (const int*)d_in[0];
    const int*   col    = (const int*)d_in[1];
    const float* weight = (const float*)d_in[2];
    const float* bias   = (const float*)d_in[3];
    float*       out    = (float*)d_out;

    const int n_edges = in_sizes[0];
    const int n4      = out_size / 4;  // float4 slices of the [N,32] output

    // 1) clear accumulator (required every call: harness does not re-zero)
    zero_kernel<<<(n4 + BLK - 1) / BLK, BLK, 0, stream>>>((float4*)out, n4);
    // 2) edge scatter-add (atomic SpMM), one edge per thread
    scatter_kernel<<<(n_edges + BLK - 1) / BLK, BLK, 0, stream>>>(row, col, weight, out, n_edges);
    // 3) bias + ReLU epilogue, in place
    bias_relu_kernel<<<(n4 + BLK - 1) / BLK, BLK, 0, stream>>>((float4*)out, bias, n4);
}